// RealAgnosticDensityInjuctedNoScaleNoBiasResidualInteractionGateBlock_21766894256495
// MI455X (gfx1250) — compile-verified
//
#include <hip/hip_runtime.h>

#define DEV __device__ __forceinline__

typedef float v2f __attribute__((ext_vector_type(2)));
typedef float v8f __attribute__((ext_vector_type(8)));
typedef unsigned int u32x4 __attribute__((ext_vector_type(4)));
typedef int i32x4 __attribute__((ext_vector_type(4)));
typedef int i32x8 __attribute__((ext_vector_type(8)));

constexpr int kN   = 8000;
constexpr int kE   = 128000;
constexpr int kC   = 128;
constexpr int kZ   = 10;
constexpr int kR   = 8;
constexpr int kEMB = 32;
constexpr int kMT  = 16;     // sum of L_DIMS (1+3+5+7)

// D = A(16x4 f32) * B(4x16 f32) + C(16x16 f32)   — CDNA5 V_WMMA_F32_16X16X4_F32
DEV v8f wmma4(v2f a, v2f b, v8f c) {
  return __builtin_amdgcn_wmma_f32_16x16x4_f32(false, a, false, b, (short)0, c,
                                               false, false);
}

DEV float silu_f(float v) { return v / (1.0f + __expf(-v)); }

#if __has_builtin(__builtin_amdgcn_tensor_load_to_lds) && \
    __has_builtin(__builtin_amdgcn_s_wait_tensorcnt)
#define USE_TDM 1
#else
#define USE_TDM 0
#endif

#if USE_TDM
// TDM gather: fetch 8 rows of `gbase` (row r = gbase + idx[r]*128 floats,
// 128 floats each) into LDS at lds_off, rows stored consecutively.
// D# layout per CDNA5 ISA §8.3-8.7 (gather mode, 32-bit indices).
// This toolchain exposes the 6-arg builtin:
//   (u32x4 g0, i32x8 g1, i32x4 g2, i32x4 g3, i32x8 extra, i32 cpol)
DEV void tdm_gather8(const float* gbase, unsigned lds_off, i32x4 idx_lo,
                     i32x4 idx_hi) {
  const unsigned long long ga = (unsigned long long)(const void*)gbase;
  u32x4 g0;
  g0[0] = 0xC0000001u;                       // count=1 | gather_mode | idx32
  g0[1] = lds_off;                            // lds_addr (bytes)
  g0[2] = (unsigned)(ga & 0xFFFFFFFFu);       // global_addr[31:0]
  g0[3] = (unsigned)((ga >> 32) & 0x01FFFFFFu) | (2u << 30);  // addr[56:32]|type=2
  i32x8 g1;
  g1[0] = (int)(2u << 16);                    // data_size=4B, mask=0
  g1[1] = (int)((unsigned)kC << 16);          // tensor_dim0=128 (bits 63:48)
  g1[2] = (int)(((unsigned)kN & 0xFFFFu) << 16);  // tensor_dim1=N (bits 95:80)
  g1[3] = (int)((unsigned)kC << 16);          // tile_dim0=128 (bits 127:112)
  g1[4] = 8;                                  // tile_dim1 = #valid indices
  g1[5] = kC;                                 // tensor_dim0_stride = 128
  g1[6] = 0;
  g1[7] = 0;
  i32x8 extra = {0, 0, 0, 0, 0, 0, 0, 0};
  __builtin_amdgcn_tensor_load_to_lds(g0, g1, idx_lo, idx_hi, extra, 0);
}
#endif

// ---------------------------------------------------------------- zero scratch
__global__ void k_zero(float* __restrict__ p, int n) {
  int i = blockIdx.x * blockDim.x + threadIdx.x;
  int stride = gridDim.x * blockDim.x;
  for (; i < n; i += stride) p[i] = 0.0f;
}

// --------------------------------------------- node GEMMs: x = nf@Wup, sc = ...
__global__ void __launch_bounds__(128) k_node(
    const float* __restrict__ nf, const float* __restrict__ na,
    const float* __restrict__ Wup, const float* __restrict__ Wskip,
    float* __restrict__ x, float* __restrict__ out_sc) {
  __shared__ float snf[4][16 * kC];   // node_feats tile
  __shared__ float sna[4][16 * kZ];   // node_attrs tile
  const int lane = threadIdx.x & 31;
  const int w    = threadIdx.x >> 5;
  const int tile = blockIdx.x * 4 + w;
  if (tile >= kN / 16) return;
  const int nb  = tile * 16;
  const int m   = lane & 15;
  const int hi  = lane >> 4;
  const int kh  = hi * 2;
  const int hi8 = hi * 8;

  // stage node_feats (16x128) and node_attrs (16x10, contiguous 160 floats)
  for (int row = 0; row < 16; ++row)
#pragma unroll
    for (int g = 0; g < 4; ++g)
      snf[w][row * kC + g * 32 + lane] =
          nf[(size_t)(nb + row) * kC + g * 32 + lane];
  for (int idx = lane; idx < 16 * kZ; idx += 32)
    sna[w][idx] = na[(size_t)nb * kZ + idx];

  // x = node_feats @ W_up  (K = 128)
  for (int ct = 0; ct < 8; ++ct) {
    v8f acc = {};
    const int col = ct * 16 + m;
    for (int k = 0; k < kC; k += 4) {
      v2f a = *(const v2f*)&snf[w][m * kC + k + kh];
      v2f b;
      b.x = Wup[(k + kh) * kC + col];
      b.y = Wup[(k + kh + 1) * kC + col];
      acc = wmma4(a, b, acc);
    }
#pragma unroll
    for (int j = 0; j < 8; ++j)
      x[(size_t)(nb + j + hi8) * kC + col] = acc[j];
  }

  // sc: A[n][p] = nf[n][p/Z] * na[n][p%Z], B = W_skip (1280 x 128), K = 1280
  for (int ct = 0; ct < 8; ++ct) {
    v8f acc = {};
    const int col = ct * 16 + m;
    for (int p = 0; p < kC * kZ; p += 4) {
      const int p0 = p + kh, p1 = p0 + 1;
      v2f a, b;
      a.x = snf[w][m * kC + p0 / kZ] * sna[w][m * kZ + p0 % kZ];
      a.y = snf[w][m * kC + p1 / kZ] * sna[w][m * kZ + p1 % kZ];
      b.x = Wskip[p0 * kC + col];
      b.y = Wskip[p1 * kC + col];
      acc = wmma4(a, b, acc);
    }
#pragma unroll
    for (int j = 0; j < 8; ++j)
      out_sc[(size_t)(nb + j + hi8) * kC + col] = acc[j];
  }
}

// ------------- fused edge kernel: MLP chain -> tp tile -> scatter, + density
__global__ void __launch_bounds__(128) k_edge(
    const float* __restrict__ ef, const float* __restrict__ eattr,
    const int* __restrict__ send, const int* __restrict__ recv,
    const float* __restrict__ W1, const float* __restrict__ W2,
    const float* __restrict__ W3, const float* __restrict__ Wtp,
    const float* __restrict__ wd, const float* __restrict__ x,
    float* __restrict__ msg, float* __restrict__ nld) {
  __shared__ float sh[4][16 * 64];    // hidden activations
  __shared__ float sxs[4][16 * kC];   // gathered x[sender] tile
  __shared__ float sea[4][16 * 16];   // edge_attrs tile
  __shared__ int   srcv[4][16];       // receiver ids
  const int lane = threadIdx.x & 31;
  const int w    = threadIdx.x >> 5;
  const int tile = blockIdx.x * 4 + w;
  if (tile >= kE / 16) return;
  const int eb  = tile * 16;
  const int m   = lane & 15;
  const int hi  = lane >> 4;
  const int kh  = hi * 2;
  const int hi8 = hi * 8;
  float* h = sh[w];

  __builtin_prefetch(Wtp, 0, 3);   // global_prefetch: W_tp panel toward WGP

#if USE_TDM
  // TDM gather-mode DMA: x[sender] rows -> sxs, overlapped with the MLP chain
  {
    i32x4 ilo, ihi;
#pragma unroll
    for (int r = 0; r < 4; ++r) ilo[r] = send[eb + r];
#pragma unroll
    for (int r = 0; r < 4; ++r) ihi[r] = send[eb + 4 + r];
    const unsigned lds0 = (unsigned)(size_t)(void*)&sxs[w][0];
    tdm_gather8(x, lds0, ilo, ihi);
#pragma unroll
    for (int r = 0; r < 4; ++r) ilo[r] = send[eb + 8 + r];
#pragma unroll
    for (int r = 0; r < 4; ++r) ihi[r] = send[eb + 12 + r];
    tdm_gather8(x, lds0 + 8u * kC * 4u, ilo, ihi);
  }
#else
  for (int row = 0; row < 16; ++row) {
    const int s = send[eb + row];
#pragma unroll
    for (int g = 0; g < 4; ++g)
      sxs[w][row * kC + g * 32 + lane] = x[(size_t)s * kC + g * 32 + lane];
  }
#endif

  // stage receiver ids and edge_attrs tile (256 contiguous floats)
  if (lane < 16) srcv[w][lane] = recv[eb + lane];
  for (int idx = lane; idx < 256; idx += 32)
    sea[w][idx] = eattr[(size_t)eb * 16 + idx];

  v8f accs[4];

  // layer 1: (16x8) @ (8x64)
#pragma unroll
  for (int ct = 0; ct < 4; ++ct) {
    v8f acc = {};
    const int col = ct * 16 + m;
#pragma unroll
    for (int k = 0; k < kR; k += 4) {
      v2f a = *(const v2f*)&ef[(size_t)(eb + m) * kR + k + kh];
      v2f b;
      b.x = W1[(k + kh) * 64 + col];
      b.y = W1[(k + kh + 1) * 64 + col];
      acc = wmma4(a, b, acc);
    }
    accs[ct] = acc;
  }
#pragma unroll
  for (int ct = 0; ct < 4; ++ct)
#pragma unroll
    for (int j = 0; j < 8; ++j)
      h[(j + hi8) * 64 + ct * 16 + m] = silu_f(accs[ct][j]);

  // layers 2 and 3: (16x64) @ (64x64), A staged via LDS
#pragma unroll
  for (int layer = 0; layer < 2; ++layer) {
    const float* W = (layer == 0) ? W2 : W3;
#pragma unroll
    for (int ct = 0; ct < 4; ++ct) {
      v8f acc = {};
      const int col = ct * 16 + m;
      for (int k = 0; k < 64; k += 4) {
        v2f a = *(const v2f*)&h[m * 64 + k + kh];
        v2f b;
        b.x = W[(k + kh) * 64 + col];
        b.y = W[(k + kh + 1) * 64 + col];
        acc = wmma4(a, b, acc);
      }
      accs[ct] = acc;
    }
#pragma unroll
    for (int ct = 0; ct < 4; ++ct)
#pragma unroll
      for (int j = 0; j < 8; ++j)
        h[(j + hi8) * 64 + ct * 16 + m] = silu_f(accs[ct][j]);
  }

  // density -> nld scatter (one edge per lane 0..15)
  if (lane < 16) {
    const int e = eb + lane;
    float d = 0.0f;
#pragma unroll
    for (int r = 0; r < kR; ++r) d += ef[e * kR + r] * wd[r];
    atomicAdd(&nld[recv[e]], tanhf(d * d));
  }

#if USE_TDM
  __builtin_amdgcn_s_wait_tensorcnt(0);   // sxs tile ready
#endif

  // tp_w tile per (l, ct), consumed immediately by the message scatter
  const int offs[4] = {0, 1, 4, 9};
  const int dls[4]  = {1, 3, 5, 7};
#pragma unroll 1
  for (int l = 0; l < 4; ++l) {
    const int offl = offs[l];
    const int dl   = dls[l];
#pragma unroll 1
    for (int ct = 0; ct < 8; ++ct) {
      v8f acc = {};
      const int col = ct * 16 + m;
      for (int k = 0; k < 64; k += 4) {
        v2f a = *(const v2f*)&h[m * 64 + k + kh];
        v2f b;
        b.x = Wtp[(k + kh) * 512 + l * kC + col];
        b.y = Wtp[(k + kh + 1) * 512 + l * kC + col];
        acc = wmma4(a, b, acc);
      }
#pragma unroll
      for (int j = 0; j < 8; ++j) {
        const int row = j + hi8;
        const int r   = srcv[w][row];
        const float v = acc[j] * sxs[w][row * kC + col];
        for (int mm = 0; mm < dl; ++mm)
          atomicAdd(&msg[(size_t)r * (kC * kMT) + col * kMT + offl + mm],
                    v * sea[w][row * 16 + offl + mm]);
      }
    }
  }
}

// ------------- final: density embedding + per-m W_lin transform + scaling
__global__ void __launch_bounds__(128) k_final(
    const float* __restrict__ msg, const float* __restrict__ nld,
    const float* __restrict__ Wlin, const float* __restrict__ Wdlin,
    float* __restrict__ out_msg) {
  __shared__ float sinv[4][16];
  __shared__ float sse[4][16 * kEMB];
  __shared__ float sA[4][16 * kC];   // holds demb tile, then staged A per mm
  const int lane = threadIdx.x & 31;
  const int w    = threadIdx.x >> 5;
  const int tile = blockIdx.x * 4 + w;
  if (tile >= kN / 16) return;
  const int nb  = tile * 16;
  const int m   = lane & 15;
  const int hi  = lane >> 4;
  const int kh  = hi * 2;
  const int hi8 = hi * 8;

  __builtin_prefetch(Wlin, 0, 3);   // global_prefetch: W_lin toward WGP

  if (lane < 16) {
    const float nl = nld[nb + lane];
    sinv[w][lane] = 1.0f / (nl + 1.0f);
    const float lg = logf(100.0f);
#pragma unroll
    for (int i = 0; i < 16; ++i) {
      const float f = __expf(-lg * (float)i / 16.0f);
      const float arg = nl * f;
      sse[w][lane * kEMB + i]      = __sinf(arg);
      sse[w][lane * kEMB + 16 + i] = __cosf(arg);
    }
  }

  // demb = sin_emb @ W_dlin  (16x32 @ 32x128) -> sA (consumed at mm==0)
  for (int ct = 0; ct < 8; ++ct) {
    v8f acc = {};
    const int col = ct * 16 + m;
#pragma unroll
    for (int k = 0; k < kEMB; k += 4) {
      v2f a = *(const v2f*)&sse[w][m * kEMB + k + kh];
      v2f b;
      b.x = Wdlin[(k + kh) * kC + col];
      b.y = Wdlin[(k + kh + 1) * kC + col];
      acc = wmma4(a, b, acc);
    }
#pragma unroll
    for (int j = 0; j < 8; ++j)
      sA[w][(j + hi8) * kC + col] = acc[j];
  }

  // out[:, :, mm] = ((msg[:, :, mm]/denom + [demb if l==0]) @ W_lin[l]) / denom
#pragma unroll 1
  for (int mm = 0; mm < kMT; ++mm) {
    const int l = (mm == 0) ? 0 : (mm < 4) ? 1 : (mm < 9) ? 2 : 3;
    const float* Wl = Wlin + (size_t)l * kC * kC;
    const bool addd = (mm == 0);

    // stage A tile (16 rows x 128 k) once per mm: msg*inv (+ demb at mm==0)
    for (int row = 0; row < 16; ++row) {
      const size_t base = (size_t)(nb + row) * (kC * kMT) + mm;
      const float iv = sinv[w][row];
#pragma unroll
      for (int g = 0; g < 4; ++g) {
        const int k = g * 32 + lane;
        float v = msg[base + (size_t)k * kMT] * iv;
        if (addd) v += sA[w][row * kC + k];   // demb (read-before-write per elem)
        sA[w][row * kC + k] = v;
      }
    }

#pragma unroll 1
    for (int ct = 0; ct < 8; ++ct) {
      v8f acc = {};
      const int col = ct * 16 + m;
      for (int k = 0; k < kC; k += 4) {
        const int k0 = k + kh;
        v2f a = *(const v2f*)&sA[w][m * kC + k0];
        v2f b;
        b.x = Wl[k0 * kC + col];   // W_lin[l][c,k]: c is the K dim
        b.y = Wl[(k0 + 1) * kC + col];
        acc = wmma4(a, b, acc);
      }
#pragma unroll
      for (int j = 0; j < 8; ++j) {
        const int row = j + hi8;
        out_msg[(size_t)(nb + row) * (kC * kMT) + col * kMT + mm] =
            acc[j] * sinv[w][row];
      }
    }
  }
}

extern "C" void kernel_launch(void* const* d_in, const int* in_sizes, int n_in,
                              void* d_out, int out_size, void* d_ws,
                              size_t ws_size, hipStream_t stream) {
  const float* node_attrs = (const float*)d_in[0];
  const float* node_feats = (const float*)d_in[1];
  const float* edge_attrs = (const float*)d_in[2];
  const float* edge_feats = (const float*)d_in[3];
  const int*   edge_index = (const int*)d_in[4];
  const float* W_up       = (const float*)d_in[5];
  const float* W_mlp1     = (const float*)d_in[6];
  const float* W_mlp2     = (const float*)d_in[7];
  const float* W_mlp3     = (const float*)d_in[8];
  const float* W_tp       = (const float*)d_in[9];
  const float* w_density  = (const float*)d_in[10];
  const float* W_skip     = (const float*)d_in[11];
  const float* W_lin      = (const float*)d_in[12];
  const float* W_dlin     = (const float*)d_in[13];

  float* out_msg = (float*)d_out;                       // (N, C, 16)
  float* out_sc  = out_msg + (size_t)kN * kC * kMT;     // (N, C)

  float* ws  = (float*)d_ws;
  float* x   = ws;                                      // N*C
  float* nld = ws + (size_t)kN * kC;                    // N
  float* msg = nld + kN;                                // N*C*16  (accumulator)

  const int* send = edge_index;
  const int* recv = edge_index + kE;

  // zero nld + msg (contiguous)
  k_zero<<<4096, 256, 0, stream>>>(nld, kN + kN * kC * kMT);

  k_node<<<(kN / 16 + 3) / 4, 128, 0, stream>>>(node_feats, node_attrs, W_up,
                                                W_skip, x, out_sc);

  k_edge<<<(kE / 16 + 3) / 4, 128, 0, stream>>>(edge_feats, edge_attrs, send,
                                                recv, W_mlp1, W_mlp2, W_mlp3,
                                                W_tp, w_density, x, msg, nld);

  k_final<<<(kN / 16 + 3) / 4, 128, 0, stream>>>(msg, nld, W_lin, W_dlin,
                                                 out_msg);
}